// WorkingMemory_3075196584325
// MI455X (gfx1250) — compile-verified
//
#include <hip/hip_runtime.h>
#include <hip/hip_bf16.h>
#include <stdint.h>

// Problem constants (match reference)
#define BSZ   2048
#define DIN   1024
#define DWM   512
#define WWIN  256
#define NH    8
#define HDIM  64
#define ASCALE 0.125f   // 1/sqrt(64)

typedef float v2f __attribute__((ext_vector_type(2)));
typedef float v4f __attribute__((ext_vector_type(4)));
typedef float v8f __attribute__((ext_vector_type(8)));
typedef unsigned int u32x4 __attribute__((ext_vector_type(4)));
typedef int i32x4 __attribute__((ext_vector_type(4)));
typedef int i32x8 __attribute__((ext_vector_type(8)));

// GEMM tiling: block 128x64, 8 waves each 32x32 (2x2 WMMA 16x16 tiles), K chunks of 32.
#define MT 128
#define NT 64
#define KT 32
#define PITCH 36   // 32 + 4 DW pad: conflict-free fragment reads; reproduced by TDM pad

#define HAVE_TDM __has_builtin(__builtin_amdgcn_tensor_load_to_lds)

#if HAVE_TDM
// Issue one TDM 2-D tile load: tile (tl1 rows x tl0 elems, 4B elems) from a
// row-major tensor with row stride str0, into LDS at ldsOff with 4-DW padding
// inserted every 32 DW (matching PITCH=36). D# packed per ISA 8.3/8.4.
// Toolchain: clang-23 lane -> 6-arg builtin (v4u, v8i, v4i, v4i, v8i, i32).
__device__ __forceinline__ void tdm_load_tile(unsigned ldsOff, const float* g,
                                              unsigned td0, unsigned td1,
                                              unsigned tl0, unsigned tl1,
                                              unsigned str0)
{
  unsigned long long ga = (unsigned long long)(uintptr_t)g;
  u32x4 g0;
  g0[0] = 1u;                                             // count=1, user D#
  g0[1] = ldsOff;                                         // lds_addr (bytes)
  g0[2] = (unsigned)ga;                                   // global_addr[31:0]
  g0[3] = (unsigned)((ga >> 32) & 0x01FFFFFFu) | (2u << 30); // [56:32] | type=2
  i32x8 g1;
  g1[0] = (int)((2u << 16)      // data_size = 4B
              | (1u << 20)      // pad_enable
              | (4u << 22)      // pad_interval: 32 DW
              | (3u << 25));    // pad_amount: 4 DW
  g1[1] = (int)((td0 & 0xFFFFu) << 16);                   // tensor_dim0[15:0]
  g1[2] = (int)((td0 >> 16) | ((td1 & 0xFFFFu) << 16));   // d0[31:16] | d1[15:0]
  g1[3] = (int)((td1 >> 16) | (tl0 << 16));               // d1[31:16] | tile_dim0
  g1[4] = (int)tl1;                                       // tile_dim1 (tile_dim2=0)
  g1[5] = (int)str0;                                      // tensor_dim0_stride[31:0]
  g1[6] = 0;                                              // stride0[47:32], stride1 lo
  g1[7] = 0;
  i32x4 gz4 = {0, 0, 0, 0};                               // 2-D tile: groups 2/3 unused
  i32x8 gz8 = {0, 0, 0, 0, 0, 0, 0, 0};                   // extra operand: zero-fill
  __builtin_amdgcn_tensor_load_to_lds(g0, g1, gz4, gz4, gz8, 0);
}
#endif

// ---------------------------------------------------------------------------
// GEMM: C[M,N] = X[M,K] @ Wt[N,K]^T + bias[N]    (torch Linear convention)
// f32 WMMA (V_WMMA_F32_16X16X4_F32) keeps full fp32 precision.
// A frag (ISA 7.12.2): lane half s, VGPR j -> K = j + 2s.  C/D: VGPR r -> M=r+8s.
// ---------------------------------------------------------------------------
__global__ __launch_bounds__(256)
void gemm_xwT(const float* __restrict__ X, const float* __restrict__ Wt,
              const float* __restrict__ bias, float* __restrict__ C,
              int M, int N, int K)
{
  const int tid  = threadIdx.x;
  const int lane = tid & 31;
  const int wave = tid >> 5;
  const int s    = lane >> 4;
  const int m    = lane & 15;
  const int wrow = wave & 3;
  const int wcol = wave >> 2;
  const int rowBase = blockIdx.y * MT;
  const int colBase = blockIdx.x * NT;

  v8f acc[2][2] = {};

#if HAVE_TDM
  // Double-buffered LDS; tiles filled by the Tensor Data Mover (async DMA),
  // so the chunk c+1 fill overlaps the 32 WMMAs of chunk c.
  __shared__ float Atile[2 * MT * PITCH];
  __shared__ float Btile[2 * NT * PITCH];

  const unsigned aOff = (unsigned)(uintptr_t)(void*)Atile;  // LDS byte offset
  const unsigned bOff = (unsigned)(uintptr_t)(void*)Btile;
  const float* Ab = X  + (size_t)rowBase * K;
  const float* Bb = Wt + (size_t)colBase * K;
  const int nChunk = K / KT;

  if (wave == 0) {
    tdm_load_tile(aOff, Ab, (unsigned)K, (unsigned)M, KT, MT, (unsigned)K);
    tdm_load_tile(bOff, Bb, (unsigned)K, (unsigned)N, KT, NT, (unsigned)K);
  }

  for (int c = 0; c < nChunk; ++c) {
    if (wave == 0) __builtin_amdgcn_s_wait_tensorcnt(0);  // chunk c landed in LDS
    __syncthreads();                                      // publish buffer c&1
    if (wave == 0 && c + 1 < nChunk) {                    // prefetch chunk c+1
      int k1 = (c + 1) * KT;
      unsigned pb = (unsigned)((c + 1) & 1);
      tdm_load_tile(aOff + pb * (MT * PITCH * 4), Ab + k1,
                    (unsigned)K, (unsigned)M, KT, MT, (unsigned)K);
      tdm_load_tile(bOff + pb * (NT * PITCH * 4), Bb + k1,
                    (unsigned)K, (unsigned)N, KT, NT, (unsigned)K);
    }
    const float* At = Atile + (c & 1) * (MT * PITCH);
    const float* Bt = Btile + (c & 1) * (NT * PITCH);
#pragma unroll
    for (int kk = 0; kk < KT; kk += 4) {
      v2f a0 = *(const v2f*)(At + (wrow * 32 +  0 + m) * PITCH + kk + 2 * s);
      v2f a1 = *(const v2f*)(At + (wrow * 32 + 16 + m) * PITCH + kk + 2 * s);
      v2f b0 = *(const v2f*)(Bt + (wcol * 32 +  0 + m) * PITCH + kk + 2 * s);
      v2f b1 = *(const v2f*)(Bt + (wcol * 32 + 16 + m) * PITCH + kk + 2 * s);
      acc[0][0] = __builtin_amdgcn_wmma_f32_16x16x4_f32(false, a0, false, b0,
                    (short)0, acc[0][0], false, false);
      acc[0][1] = __builtin_amdgcn_wmma_f32_16x16x4_f32(false, a0, false, b1,
                    (short)0, acc[0][1], false, false);
      acc[1][0] = __builtin_amdgcn_wmma_f32_16x16x4_f32(false, a1, false, b0,
                    (short)0, acc[1][0], false, false);
      acc[1][1] = __builtin_amdgcn_wmma_f32_16x16x4_f32(false, a1, false, b1,
                    (short)0, acc[1][1], false, false);
    }
    __syncthreads();   // compute done before TDM may overwrite this buffer
  }
#else
  // Fallback: cooperative fill, loads staged in registers before LDS stores.
  __shared__ float Atile[MT * PITCH];
  __shared__ float Btile[NT * PITCH];

  for (int k0 = 0; k0 < K; k0 += KT) {
    v4f ar[4], br[2];
#pragma unroll
    for (int u = 0; u < 4; ++u) {
      int fi = tid + u * 256, r = fi >> 3, q = fi & 7;
      ar[u] = *(const v4f*)(X + (size_t)(rowBase + r) * K + k0 + q * 4);
    }
#pragma unroll
    for (int u = 0; u < 2; ++u) {
      int fi = tid + u * 256, n = fi >> 3, q = fi & 7;
      br[u] = *(const v4f*)(Wt + (size_t)(colBase + n) * K + k0 + q * 4);
    }
#pragma unroll
    for (int u = 0; u < 4; ++u) {
      int fi = tid + u * 256, r = fi >> 3, q = fi & 7;
      *(v4f*)(Atile + r * PITCH + q * 4) = ar[u];
    }
#pragma unroll
    for (int u = 0; u < 2; ++u) {
      int fi = tid + u * 256, n = fi >> 3, q = fi & 7;
      *(v4f*)(Btile + n * PITCH + q * 4) = br[u];
    }
    __syncthreads();
#pragma unroll
    for (int kk = 0; kk < KT; kk += 4) {
      v2f a0 = *(const v2f*)(Atile + (wrow * 32 +  0 + m) * PITCH + kk + 2 * s);
      v2f a1 = *(const v2f*)(Atile + (wrow * 32 + 16 + m) * PITCH + kk + 2 * s);
      v2f b0 = *(const v2f*)(Btile + (wcol * 32 +  0 + m) * PITCH + kk + 2 * s);
      v2f b1 = *(const v2f*)(Btile + (wcol * 32 + 16 + m) * PITCH + kk + 2 * s);
      acc[0][0] = __builtin_amdgcn_wmma_f32_16x16x4_f32(false, a0, false, b0,
                    (short)0, acc[0][0], false, false);
      acc[0][1] = __builtin_amdgcn_wmma_f32_16x16x4_f32(false, a0, false, b1,
                    (short)0, acc[0][1], false, false);
      acc[1][0] = __builtin_amdgcn_wmma_f32_16x16x4_f32(false, a1, false, b0,
                    (short)0, acc[1][0], false, false);
      acc[1][1] = __builtin_amdgcn_wmma_f32_16x16x4_f32(false, a1, false, b1,
                    (short)0, acc[1][1], false, false);
    }
    __syncthreads();
  }
#endif

  // Epilogue: D VGPR r / half s -> row r + 8s, col = lane%16
#pragma unroll
  for (int i = 0; i < 2; ++i) {
#pragma unroll
    for (int j = 0; j < 2; ++j) {
      int col = colBase + wcol * 32 + j * 16 + m;
      float bcol = bias[col];
#pragma unroll
      for (int r = 0; r < 8; ++r) {
        int row = rowBase + wrow * 32 + i * 16 + 8 * s + r;
        C[(size_t)row * N + col] = acc[i][j][r] + bcol;
      }
    }
  }
}

// ---------------------------------------------------------------------------
// Attention over the circular window. One block per (batch, head).
// The ptr row is substituted from the freshly computed k/v (never materializes
// the updated 1 GB K/V). Streams wm_K/wm_V non-temporally: 2.15 GB at
// 23.3 TB/s ~= 92 us — this kernel is the roofline-dominant hot spot.
// ---------------------------------------------------------------------------
__global__ __launch_bounds__(256)
void attn_window(const float* __restrict__ qbuf, const float* __restrict__ kbuf,
                 const float* __restrict__ vbuf, const float* __restrict__ wmK,
                 const float* __restrict__ wmV,
                 const unsigned char* __restrict__ wmValid,
                 const unsigned char* __restrict__ resetMask,
                 const int* __restrict__ wmPtr, float* __restrict__ outBuf)
{
  __shared__ float q_s[HDIM];
  __shared__ float p_s[WWIN];
  __shared__ float red[WWIN];

  const int b = blockIdx.x / NH;
  const int h = blockIdx.x % NH;
  const int tid = threadIdx.x;

  const bool reset = resetMask[b] != 0;
  const int  ptr   = reset ? 0 : wmPtr[b];
  const size_t headOff = (size_t)b * DWM + h * HDIM;

  if (tid < HDIM) q_s[tid] = qbuf[headOff + tid];
  __syncthreads();

  // ---- scores: thread w computes q . K[w] over 64 dims ----
  const int w = tid;
  const bool atPtr = (w == ptr);
  const float* krow = atPtr ? (kbuf + headOff)
                            : (wmK + ((size_t)b * WWIN + w) * DWM + h * HDIM);
  const bool valid = atPtr || (!reset && (wmValid[(size_t)b * WWIN + w] != 0));

  float sacc = 0.f;
#pragma unroll
  for (int dq = 0; dq < HDIM / 4; ++dq) {
    v4f kv = __builtin_nontemporal_load((const v4f*)krow + dq);
    v4f qv = *((const v4f*)q_s + dq);
    sacc += kv.x * qv.x + kv.y * qv.y + kv.z * qv.z + kv.w * qv.w;
  }
  float score = valid ? sacc * ASCALE : -__builtin_inff();

  // ---- block softmax (max then sum) ----
  red[tid] = score;
  __syncthreads();
  for (int st = 128; st > 0; st >>= 1) {
    if (tid < st) red[tid] = fmaxf(red[tid], red[tid + st]);
    __syncthreads();
  }
  float mx = red[0];
  __syncthreads();
  float e = valid ? __expf(score - mx) : 0.f;
  red[tid] = e;
  __syncthreads();
  for (int st = 128; st > 0; st >>= 1) {
    if (tid < st) red[tid] += red[tid + st];
    __syncthreads();
  }
  float ssum = red[0];
  float p = (ssum > 0.f) ? (e / ssum) : 0.f;   // nan_to_num semantics
  p_s[tid] = p;
  __syncthreads();

  // ---- out[d] = sum_w p[w] * V[w,d]; thread=(d, quarter g); coalesced ----
  const int d = tid & 63;
  const int g = tid >> 6;            // wave-uniform -> scalar branch below
  float oacc = 0.f;
  for (int wi = g * 64; wi < g * 64 + 64; ++wi) {
    float pv = p_s[wi];
    float vv = (wi == ptr)
                 ? vbuf[headOff + d]
                 : __builtin_nontemporal_load(
                       wmV + ((size_t)b * WWIN + wi) * DWM + h * HDIM + d);
    oacc += pv * vv;
  }
  red[tid] = oacc;
  __syncthreads();
  if (tid < 64) {
    float o = red[tid] + red[tid + 64] + red[tid + 128] + red[tid + 192];
    outBuf[headOff + tid] = o;
  }
}

// ---------------------------------------------------------------------------
extern "C" void kernel_launch(void* const* d_in, const int* in_sizes, int n_in,
                              void* d_out, int out_size, void* d_ws, size_t ws_size,
                              hipStream_t stream)
{
  (void)in_sizes; (void)n_in; (void)out_size; (void)ws_size;

  const float*         x       = (const float*)d_in[0];
  const unsigned char* reset   = (const unsigned char*)d_in[1];
  const float*         wmK     = (const float*)d_in[2];
  const float*         wmV     = (const float*)d_in[3];
  const unsigned char* wmValid = (const unsigned char*)d_in[4];
  const int*           wmPtr   = (const int*)d_in[5];
  const float*         Wq      = (const float*)d_in[6];
  const float*         bq      = (const float*)d_in[7];
  const float*         Wk      = (const float*)d_in[8];
  const float*         bk      = (const float*)d_in[9];
  const float*         Wv      = (const float*)d_in[10];
  const float*         bv      = (const float*)d_in[11];
  const float*         Wo      = (const float*)d_in[12];
  const float*         bo      = (const float*)d_in[13];
  float*               yout    = (float*)d_out;

  float* ws = (float*)d_ws;
  const size_t QKVN = (size_t)BSZ * DWM;   // 1 Mi floats each
  float* qb = ws;
  float* kb = ws + QKVN;
  float* vb = ws + 2 * QKVN;
  float* ab = ws + 3 * QKVN;

  dim3 blk(256);
  dim3 gQKV(DWM / NT, BSZ / MT);           // (8, 16)
  gemm_xwT<<<gQKV, blk, 0, stream>>>(x, Wq, bq, qb, BSZ, DWM, DIN);
  gemm_xwT<<<gQKV, blk, 0, stream>>>(x, Wk, bk, kb, BSZ, DWM, DIN);
  gemm_xwT<<<gQKV, blk, 0, stream>>>(x, Wv, bv, vb, BSZ, DWM, DIN);

  attn_window<<<BSZ * NH, blk, 0, stream>>>(qb, kb, vb, wmK, wmV, wmValid,
                                            reset, wmPtr, ab);

  dim3 gO(DIN / NT, BSZ / MT);             // (16, 16)
  gemm_xwT<<<gO, blk, 0, stream>>>(ab, Wo, bo, yout, BSZ, DIN, DWM);
}